// GraphAttentionConv_14766097563783
// MI455X (gfx1250) — compile-verified
//
#include <hip/hip_runtime.h>
#include <hip/hip_bf16.h>

#define N_NODES 8192
#define F_DIM   256
#define MAXN    1024   // neighbor cap (avg ~34, 4.5-sigma tail ~60)

typedef __attribute__((ext_vector_type(2))) float v2f;
typedef __attribute__((ext_vector_type(8))) float v8f;
typedef __attribute__((ext_vector_type(4))) int   v4i;

typedef __attribute__((address_space(1))) v4i gv4i;   // global int4
typedef __attribute__((address_space(3))) v4i lv4i;   // LDS int4

__device__ __forceinline__ float leaky01(float x) {
    return x > 0.0f ? x : 0.01f * x;
}

// --- CDNA5 async global->LDS helpers ---------------------------------------
__device__ __forceinline__ void async_copy_b128(const float* g, float* l) {
#if __has_builtin(__builtin_amdgcn_global_load_async_to_lds_b128)
    // AS1 address = plain 64-bit VA; AS3 address = low 32 bits of generic
    // LDS pointer.
    gv4i* gp = (gv4i*)(unsigned long long)(uintptr_t)g;
    lv4i* lp = (lv4i*)(unsigned int)(uintptr_t)l;
    __builtin_amdgcn_global_load_async_to_lds_b128(gp, lp, /*offset=*/0, /*cpol=*/0);
#else
    *(float4*)l = *(const float4*)g;        // synchronous fallback
#endif
}

template <int Ncnt>
__device__ __forceinline__ void wait_asynccnt() {
#if __has_builtin(__builtin_amdgcn_s_wait_asynccnt)
    __builtin_amdgcn_s_wait_asynccnt((unsigned short)Ncnt);
    asm volatile("" ::: "memory");   // compiler barrier: keep LDS reads below
#else
    if      constexpr (Ncnt == 0) asm volatile("s_wait_asynccnt 0x0" ::: "memory");
    else if constexpr (Ncnt == 1) asm volatile("s_wait_asynccnt 0x1" ::: "memory");
    else if constexpr (Ncnt == 2) asm volatile("s_wait_asynccnt 0x2" ::: "memory");
    else                          asm volatile("s_wait_asynccnt 0x3" ::: "memory");
#endif
}

// ---------------------------------------------------------------------------
// Kernel 1: Xp = X @ W^T + b  via V_WMMA_F32_16X16X4_F32 (full fp32 path).
// One wave per 16x16 output tile; block = 512 threads = 16 waves = one
// 16-row x 256-col stripe. 64 WMMA ops per wave over K=256.
// ---------------------------------------------------------------------------
__global__ __launch_bounds__(512) void gat_gemm_wmma(
    const float* __restrict__ X,     // [N, F]
    const float* __restrict__ W,     // [F_out, F_in]
    const float* __restrict__ bias,  // [F_out]
    float* __restrict__ Xp)          // [N, F]
{
    const int lane = threadIdx.x & 31;
    const int wave = threadIdx.x >> 5;          // 0..15 -> column tile
    const int row0 = blockIdx.x * 16;
    const int col0 = wave * 16;

    const int mn = lane & 15;                   // M for A-frag, N for B-frag
    const int kb = (lane >> 4) * 2;             // K sub-offset per ISA layout

    const float* Arow = X + (size_t)(row0 + mn) * F_DIM + kb;
    const float* Brow = W + (size_t)(col0 + mn) * F_DIM + kb;

    v8f c = {};
    #pragma unroll 4
    for (int k = 0; k < F_DIM; k += 4) {
        v2f a; a.x = Arow[k]; a.y = Arow[k + 1];
        v2f b; b.x = Brow[k]; b.y = Brow[k + 1];
        c = __builtin_amdgcn_wmma_f32_16x16x4_f32(
                false, a, false, b, (short)0, c, false, false);
    }

    // D layout: VGPR r -> (M = (lane>=16 ? 8 : 0)+r, N = lane&15)
    const int mbase = (lane >> 4) * 8;
    const float bn = bias[col0 + mn];
    #pragma unroll
    for (int r = 0; r < 8; ++r) {
        Xp[(size_t)(row0 + mbase + r) * F_DIM + col0 + mn] = c[r] + bn;
    }
}

// ---------------------------------------------------------------------------
// Kernel 2: a_src[i] = Xp[i,:].S[:F],  a_dst[i] = Xp[i,:].S[F:]
// One wave per row, wave32 shuffle reduction.
// ---------------------------------------------------------------------------
__global__ __launch_bounds__(256) void gat_scores(
    const float* __restrict__ Xp,
    const float* __restrict__ S,
    float* __restrict__ a_src,
    float* __restrict__ a_dst)
{
    const int lane = threadIdx.x & 31;
    const int wave = threadIdx.x >> 5;
    const int row  = blockIdx.x * 8 + wave;

    const float* xr = Xp + (size_t)row * F_DIM;
    float s0 = 0.0f, s1 = 0.0f;
    #pragma unroll
    for (int p = 0; p < F_DIM / 32; ++p) {
        const int f = lane + 32 * p;
        const float v = xr[f];
        s0 += v * S[f];
        s1 += v * S[F_DIM + f];
    }
    #pragma unroll
    for (int off = 16; off > 0; off >>= 1) {
        s0 += __shfl_xor(s0, off, 32);
        s1 += __shfl_xor(s1, off, 32);
    }
    if (lane == 0) { a_src[row] = s0; a_dst[row] = s1; }
}

// ---------------------------------------------------------------------------
// Kernel 3: per-row neighbor scan + segment softmax + sparse aggregate.
// One 256-thread block (8 waves) per row. The 268 MB adj stream is the
// bottleneck; stage it through LDS with async b128 copies (ASYNCcnt),
// quad-buffered per wave, overlapping DMA with predicate/append VALU work.
// ---------------------------------------------------------------------------
#define DEPTH        4      // ring depth (async ops in flight per wave)
#define WAVE_COLS    1024   // columns owned by each wave (8 waves * 1024 = 8192)
#define CHUNK_FLOATS 128    // 32 lanes * float4 per chunk (512 B, one async op)
#define NUM_CHUNKS   (WAVE_COLS / CHUNK_FLOATS)   // 8

__global__ __launch_bounds__(256) void gat_aggregate(
    const float* __restrict__ adj,    // [N, N]
    const float* __restrict__ Xp,     // [N, F]
    const float* __restrict__ a_src,  // [N]
    const float* __restrict__ a_dst,  // [N]
    float* __restrict__ out)          // [N, F]
{
    const int row = blockIdx.x;
    const int tid = threadIdx.x;
    const int wv  = tid >> 5;
    const int ln  = tid & 31;

    __shared__ int   nbr[MAXN];
    __shared__ float wts[MAXN];
    __shared__ int   cnt;
    __shared__ float red[256];
    __shared__ __align__(16) float stage[8 * DEPTH * CHUNK_FLOATS];  // 16 KB

    if (tid == 0) cnt = 0;
    __syncthreads();

    // --- Pass A: async-stream adj row through LDS, collect neighbors -----
    const float* gsrc   = adj + (size_t)row * N_NODES + wv * WAVE_COLS;
    float*       mystg  = stage + wv * DEPTH * CHUNK_FLOATS;
    const int    lnoff  = ln * 4;

    #pragma unroll
    for (int c = 0; c < DEPTH; ++c)
        async_copy_b128(gsrc + c * CHUNK_FLOATS + lnoff,
                        mystg + c * CHUNK_FLOATS + lnoff);

    #pragma unroll
    for (int c = 0; c < NUM_CHUNKS; ++c) {
        // in-order completion: chunk c resident once outstanding <= pending-after-c
        if      (c <= NUM_CHUNKS - DEPTH)     wait_asynccnt<DEPTH - 1>();
        else if (c == NUM_CHUNKS - DEPTH + 1) wait_asynccnt<DEPTH - 2>();
        else if (c == NUM_CHUNKS - DEPTH + 2) wait_asynccnt<DEPTH - 3>();
        else                                  wait_asynccnt<0>();

        const float4 v = *(const float4*)(mystg + (c % DEPTH) * CHUNK_FLOATS + lnoff);
        const int base = wv * WAVE_COLS + c * CHUNK_FLOATS + lnoff;
        if (v.x != 0.0f || base + 0 == row) { int s = atomicAdd(&cnt, 1); if (s < MAXN) nbr[s] = base + 0; }
        if (v.y != 0.0f || base + 1 == row) { int s = atomicAdd(&cnt, 1); if (s < MAXN) nbr[s] = base + 1; }
        if (v.z != 0.0f || base + 2 == row) { int s = atomicAdd(&cnt, 1); if (s < MAXN) nbr[s] = base + 2; }
        if (v.w != 0.0f || base + 3 == row) { int s = atomicAdd(&cnt, 1); if (s < MAXN) nbr[s] = base + 3; }

        const int nc = c + DEPTH;
        if (nc < NUM_CHUNKS)
            async_copy_b128(gsrc + nc * CHUNK_FLOATS + lnoff,
                            mystg + (nc % DEPTH) * CHUNK_FLOATS + lnoff);
    }
    __syncthreads();
    const int n = cnt < MAXN ? cnt : MAXN;

    // --- Row max: leaky_relu monotone -> max_j e_ij at max_j a_dst[j] -----
    float m = -3.0e38f;
    for (int k = tid; k < n; k += 256) m = fmaxf(m, a_dst[nbr[k]]);
    red[tid] = m;
    __syncthreads();
    #pragma unroll
    for (int s = 128; s > 0; s >>= 1) {
        if (tid < s) red[tid] = fmaxf(red[tid], red[tid + s]);
        __syncthreads();
    }
    const float asrc = a_src[row];
    const float emax = leaky01(asrc + red[0]);
    __syncthreads();

    // --- Softmax numerators ----------------------------------------------
    for (int k = tid; k < n; k += 256) {
        const float e = leaky01(asrc + a_dst[nbr[k]]);
        wts[k] = __expf(e - emax);
    }
    __syncthreads();

    // --- Denominator -----------------------------------------------------
    float d = 0.0f;
    for (int k = tid; k < n; k += 256) d += wts[k];
    red[tid] = d;
    __syncthreads();
    #pragma unroll
    for (int s = 128; s > 0; s >>= 1) {
        if (tid < s) red[tid] += red[tid + s];
        __syncthreads();
    }
    const float denom = red[0];

    // --- Aggregate: thread t owns feature t; coalesced Xp row reads ------
    float acc = 0.0f;
    for (int k = 0; k < n; ++k) {
        acc += wts[k] * Xp[(size_t)nbr[k] * F_DIM + tid];
    }
    const float z = acc / denom;
    out[(size_t)row * F_DIM + tid] = 1.0f / (1.0f + __expf(-z));
}

// ---------------------------------------------------------------------------
extern "C" void kernel_launch(void* const* d_in, const int* in_sizes, int n_in,
                              void* d_out, int out_size, void* d_ws, size_t ws_size,
                              hipStream_t stream) {
    const float* X   = (const float*)d_in[0];   // [8192, 256]
    const float* adj = (const float*)d_in[1];   // [8192, 8192]
    const float* W   = (const float*)d_in[2];   // [256, 256]
    const float* b   = (const float*)d_in[3];   // [256]
    const float* S   = (const float*)d_in[4];   // [512]
    float* out = (float*)d_out;                 // [8192, 256]

    float* Xp    = (float*)d_ws;                          // 8 MB
    float* a_src = Xp + (size_t)N_NODES * F_DIM;          // 32 KB
    float* a_dst = a_src + N_NODES;                       // 32 KB

    gat_gemm_wmma<<<N_NODES / 16, 512, 0, stream>>>(X, W, b, Xp);
    gat_scores<<<N_NODES / 8, 256, 0, stream>>>(Xp, S, a_src, a_dst);
    gat_aggregate<<<N_NODES, 256, 0, stream>>>(adj, Xp, a_src, a_dst, out);
}